// SpatialTemporalLayer_46084999086094
// MI455X (gfx1250) — compile-verified
//
#include <hip/hip_runtime.h>

typedef __attribute__((ext_vector_type(2))) float v2f;
typedef __attribute__((ext_vector_type(8))) float v8f;

#define Bv 4
#define Tv 32
#define Nv 1000
#define Cv 64
#define Ev 16000
#define BT (Bv * Tv)              // 128
#define ROWS (Bv * Tv * Nv)       // 128000
#define ELEMS (ROWS * Cv)         // 8192000
#define EPSV 1e-5f
#define CONV_K (9 * Cv)           // 576
#define CONV_KSTEPS (CONV_K / 4)  // 144
#define LPITCH 66                 // LDS row pitch (floats): 8B-aligned rows, low conflicts

// ---------------------------------------------------------------------------
// 1) init: zero int degree counts, bucket cursors, BN stat accumulators
// ---------------------------------------------------------------------------
__global__ void init_kernel(int* __restrict__ degi, int* __restrict__ curs,
                            float* __restrict__ sums) {
    int i = blockIdx.x * blockDim.x + threadIdx.x;
    if (i < Nv) { degi[i] = 0; curs[i] = 0; }
    if (i < 4 * Cv) sums[i] = 0.0f;
}

// ---------------------------------------------------------------------------
// 2) degree count over destination nodes
// ---------------------------------------------------------------------------
__global__ void deg_kernel(const long long* __restrict__ ei, int* __restrict__ degi) {
    int e = blockIdx.x * blockDim.x + threadIdx.x;
    if (e < Ev) atomicAdd(&degi[(int)ei[Ev + e]], 1);
}

// ---------------------------------------------------------------------------
// 3) exclusive scan over N=1000 nodes + dinv = rsqrt(deg+1)  (trivial size)
// ---------------------------------------------------------------------------
__global__ void scan_kernel(const int* __restrict__ degi, int* __restrict__ offs,
                            float* __restrict__ dinv) {
    int acc = 0;
    for (int n = 0; n < Nv; ++n) {
        offs[n] = acc;
        acc += degi[n];
        dinv[n] = rsqrtf((float)(degi[n] + 1));  // +1 self loop
    }
    offs[Nv] = acc;
}

// ---------------------------------------------------------------------------
// 4) bucket fill: CSR by destination (src indices)
// ---------------------------------------------------------------------------
__global__ void fill_kernel(const long long* __restrict__ ei, const int* __restrict__ offs,
                            int* __restrict__ curs, int* __restrict__ srcs) {
    int e = blockIdx.x * blockDim.x + threadIdx.x;
    if (e >= Ev) return;
    int d = (int)ei[Ev + e];
    int pos = atomicAdd(&curs[d], 1);
    srcs[offs[d] + pos] = (int)ei[e];
}

// ---------------------------------------------------------------------------
// 5) reorder W and conv_w into WMMA B-operand per-lane order:
//    one float2 per (tile, kstep, lane) -> coalesced global_load_b64 in GEMMs
// ---------------------------------------------------------------------------
__global__ void reorder_w_kernel(const float* __restrict__ W, const float* __restrict__ cw,
                                 float* __restrict__ Wb, float* __restrict__ Cwb) {
    int i = blockIdx.x * blockDim.x + threadIdx.x;
    if (i < 4 * 16 * 32) {  // Wb: ct(4) x kk4(16) x lane(32)
        int lane = i & 31, kk4 = (i >> 5) & 15, ct = i >> 9;
        int hi = lane >> 4, lo = lane & 15;
        int k0 = kk4 * 4 + 2 * hi;
        Wb[2 * i + 0] = W[(k0 + 0) * Cv + ct * 16 + lo];
        Wb[2 * i + 1] = W[(k0 + 1) * Cv + ct * 16 + lo];
    }
    if (i < 4 * CONV_KSTEPS * 32) {  // Cwb: ot(4) x kk4(144) x lane(32)
        int lane = i & 31, kk4 = (i >> 5) % CONV_KSTEPS, ot = i / (CONV_KSTEPS * 32);
        int hi = lane >> 4, lo = lane & 15;
        int o = ot * 16 + lo;
        int k0 = kk4 * 4 + 2 * hi;
        int tap = k0 >> 6, ci = k0 & 63;
        Cwb[2 * i + 0] = cw[(o * Cv + ci) * 9 + tap];
        int k1 = k0 + 1;
        tap = k1 >> 6; ci = k1 & 63;
        Cwb[2 * i + 1] = cw[(o * Cv + ci) * 9 + tap];
    }
}

// ---------------------------------------------------------------------------
// 6) xw = x @ W via V_WMMA_F32_16X16X4_F32; B from pre-swizzled Wb
// ---------------------------------------------------------------------------
__global__ __launch_bounds__(128) void gemm_xw_kernel(const float* __restrict__ x,
                                                      const float* __restrict__ Wb,
                                                      float* __restrict__ xw) {
    const int ct = threadIdx.x >> 5;     // wave -> column tile
    const int lane = threadIdx.x & 31;
    const int hi = lane >> 4;
    const int lo = lane & 15;
    const int rowbase = blockIdx.x * 16;

    v8f acc = {};
    const float* xrow = x + (long long)(rowbase + lo) * Cv;
    const float* wb = Wb + ((ct * 16) * 32 + lane) * 2;
#pragma unroll
    for (int kk4 = 0; kk4 < 16; ++kk4) {
        v2f a = *(const v2f*)(xrow + kk4 * 4 + 2 * hi);
        v2f bm = *(const v2f*)(wb + kk4 * 64);
        acc = __builtin_amdgcn_wmma_f32_16x16x4_f32(false, a, false, bm,
                                                    (short)0, acc, false, false);
    }
#pragma unroll
    for (int r = 0; r < 8; ++r)
        xw[(long long)(rowbase + r + 8 * hi) * Cv + ct * 16 + lo] = acc[r];
}

// ---------------------------------------------------------------------------
// 7) GCN aggregation (gather, no atomics) + GCN bias fused:
//    so = selfloop + sum over incident edges + bias
// ---------------------------------------------------------------------------
__global__ void agg_gather_kernel(const float* __restrict__ xw, const int* __restrict__ offs,
                                  const int* __restrict__ srcs, const float* __restrict__ dinv,
                                  const float* __restrict__ bias, float* __restrict__ so) {
    int idx = blockIdx.x * blockDim.x + threadIdx.x;
    const int total = BT * Nv * 16;
    if (idx >= total) return;
    int cq = idx & 15;
    int t2 = idx >> 4;
    int n = t2 % Nv;
    int bt = t2 / Nv;

    const float4* xw4 = (const float4*)xw;
    float dn = dinv[n];
    float4 v = xw4[(long long)(bt * Nv + n) * 16 + cq];
    float w = dn * dn;  // self-loop norm
    float4 acc;
    acc.x = v.x * w; acc.y = v.y * w; acc.z = v.z * w; acc.w = v.w * w;

    int j0 = offs[n], j1 = offs[n + 1];
    for (int j = j0; j < j1; ++j) {
        int s = srcs[j];
        float ws = dinv[s] * dn;
        float4 u = xw4[(long long)(bt * Nv + s) * 16 + cq];
        acc.x += u.x * ws; acc.y += u.y * ws; acc.z += u.z * ws; acc.w += u.w * ws;
    }
    const float4* b4 = (const float4*)bias;
    float4 bb = b4[cq];
    acc.x += bb.x; acc.y += bb.y; acc.z += bb.z; acc.w += bb.w;
    ((float4*)so)[(long long)(bt * Nv + n) * 16 + cq] = acc;
}

// ---------------------------------------------------------------------------
// 8) per-channel sum/sumsq (read-only; used for both BN1 and BN2)
// ---------------------------------------------------------------------------
__global__ __launch_bounds__(64) void stats_kernel(const float* __restrict__ buf,
                                                   float* __restrict__ sum,
                                                   float* __restrict__ sumsq) {
    int c = threadIdx.x;
    float s = 0.0f, sq = 0.0f;
    for (long long r = blockIdx.x; r < ROWS; r += gridDim.x) {
        float v = buf[r * Cv + c];
        s += v;
        sq += v * v;
    }
    atomicAdd(&sum[c], s);
    atomicAdd(&sumsq[c], sq);
}

// ---------------------------------------------------------------------------
// 9) temporal conv as WMMA GEMM. BN1+ReLU applied on the fly while staging
//    the A slice (24 t-rows x 64 ch, zero-padded halo) into LDS.
//    Block = (b, n, ttile); 4 waves = 4 o-tiles; K = 576 (kglob = tap*64+ci).
// ---------------------------------------------------------------------------
__global__ __launch_bounds__(128) void conv_kernel(const float* __restrict__ so,
                                                   const float* __restrict__ sums,
                                                   const float* __restrict__ g1,
                                                   const float* __restrict__ b1,
                                                   const float* __restrict__ Cwb,
                                                   const float* __restrict__ cb,
                                                   float* __restrict__ cv) {
    __shared__ float s_z[24 * LPITCH];

    int blk = blockIdx.x;
    int ttile = blk & 1;
    int bn = blk >> 1;
    int n = bn % Nv;
    int b = bn / Nv;
    const int wave = threadIdx.x >> 5;
    const int lane = threadIdx.x & 31;
    const int hi = lane >> 4;
    const int lo = lane & 15;

    // BN1 constants for this thread's staging channel (constant across strides)
    const float inv_cnt = 1.0f / (float)ROWS;
    {
        int c = threadIdx.x & 63;
        float mean = sums[c] * inv_cnt;
        float var = sums[Cv + c] * inv_cnt - mean * mean;
        float is = rsqrtf(var + EPSV);
        float g = g1[c], bb = b1[c];
        const int tbase = ttile * 16 - 4;
        for (int i = threadIdx.x; i < 24 * 64; i += 128) {
            int tl = i >> 6, ci = i & 63;  // ci == c
            int ti = tbase + tl;
            float v = 0.0f;
            if (ti >= 0 && ti < Tv) {
                float u = so[(((long long)b * Tv + ti) * Nv + n) * Cv + ci];
                v = (u - mean) * is * g + bb;
                v = v > 0.0f ? v : 0.0f;
            }
            s_z[tl * LPITCH + ci] = v;
        }
    }
    __syncthreads();

    v8f acc = {};
    const float* cwb = Cwb + ((wave * CONV_KSTEPS) * 32 + lane) * 2;
#pragma unroll
    for (int tap = 0; tap < 9; ++tap) {
        const float* row = s_z + (lo + tap) * LPITCH + 2 * hi;   // 8B aligned
        const float* cwt = cwb + (tap * 16) * 64;
#pragma unroll
        for (int kk = 0; kk < 16; ++kk) {
            v2f a = *(const v2f*)(row + kk * 4);     // ds_load_b64
            v2f bm = *(const v2f*)(cwt + kk * 64);   // global_load_b64
            acc = __builtin_amdgcn_wmma_f32_16x16x4_f32(false, a, false, bm,
                                                        (short)0, acc, false, false);
        }
    }
    float bo = cb[wave * 16 + lo];
#pragma unroll
    for (int r = 0; r < 8; ++r) {
        int t = ttile * 16 + r + 8 * hi;
        cv[(((long long)b * Tv + t) * Nv + n) * Cv + wave * 16 + lo] = acc[r] + bo;
    }
}

// ---------------------------------------------------------------------------
// 10) out = relu(BN2(cv) + x)
// ---------------------------------------------------------------------------
__global__ void final_kernel(const float* __restrict__ cv, const float* __restrict__ x,
                             const float* __restrict__ sums, const float* __restrict__ g2,
                             const float* __restrict__ b2, float* __restrict__ out) {
    int i = blockIdx.x * blockDim.x + threadIdx.x;
    if (i >= ELEMS) return;
    int c = i & 63;
    const float inv_cnt = 1.0f / (float)ROWS;
    float mean = sums[2 * Cv + c] * inv_cnt;
    float var = sums[3 * Cv + c] * inv_cnt - mean * mean;
    float is = rsqrtf(var + EPSV);
    float v = (cv[i] - mean) * is * g2[c] + b2[c] + x[i];
    out[i] = v > 0.0f ? v : 0.0f;
}

// ---------------------------------------------------------------------------
extern "C" void kernel_launch(void* const* d_in, const int* in_sizes, int n_in,
                              void* d_out, int out_size, void* d_ws, size_t ws_size,
                              hipStream_t stream) {
    const float* x = (const float*)d_in[0];
    const long long* ei = (const long long*)d_in[1];
    const float* W = (const float*)d_in[2];
    const float* bias = (const float*)d_in[3];
    const float* g1 = (const float*)d_in[4];
    const float* b1 = (const float*)d_in[5];
    const float* cw = (const float*)d_in[6];
    const float* cb = (const float*)d_in[7];
    const float* g2 = (const float*)d_in[8];
    const float* b2 = (const float*)d_in[9];
    float* out = (float*)d_out;

    // workspace layout
    float* ws = (float*)d_ws;
    float* buf0 = ws;                         // ELEMS: xw, later cv
    float* buf1 = ws + (size_t)ELEMS;         // ELEMS: so (agg + bias)
    float* dinv = ws + 2 * (size_t)ELEMS;     // 1024
    float* sums = dinv + 1024;                // 256: [sum1|sumsq1|sum2|sumsq2]
    float* Wb = sums + 256;                   // 4096
    float* Cwb = Wb + 4096;                   // 36864
    int* degi = (int*)(Cwb + 36864);          // 1024
    int* offs = degi + 1024;                  // 1024 (need Nv+1)
    int* curs = offs + 1024;                  // 1024
    int* srcs = curs + 1024;                  // Ev

    init_kernel<<<4, 256, 0, stream>>>(degi, curs, sums);
    deg_kernel<<<(Ev + 255) / 256, 256, 0, stream>>>(ei, degi);
    scan_kernel<<<1, 1, 0, stream>>>(degi, offs, dinv);
    fill_kernel<<<(Ev + 255) / 256, 256, 0, stream>>>(ei, offs, curs, srcs);
    reorder_w_kernel<<<(4 * CONV_KSTEPS * 32 + 255) / 256, 256, 0, stream>>>(W, cw, Wb, Cwb);
    gemm_xw_kernel<<<ROWS / 16, 128, 0, stream>>>(x, Wb, buf0 /* xw */);
    agg_gather_kernel<<<(BT * Nv * 16 + 255) / 256, 256, 0, stream>>>(buf0, offs, srcs, dinv,
                                                                      bias, buf1 /* so */);
    stats_kernel<<<512, 64, 0, stream>>>(buf1 /* so */, sums, sums + Cv);
    conv_kernel<<<Bv * Nv * 2, 128, 0, stream>>>(buf1 /* so */, sums, g1, b1, Cwb, cb,
                                                 buf0 /* cv */);
    stats_kernel<<<512, 64, 0, stream>>>(buf0 /* cv */, sums + 2 * Cv, sums + 3 * Cv);
    final_kernel<<<ELEMS / 256, 256, 0, stream>>>(buf0 /* cv */, x, sums, g2, b2, out);
}